// MileCutLoss_41120016892019
// MI455X (gfx1250) — compile-verified
//
#include <hip/hip_runtime.h>
#include <cstdint>
#include <cstddef>

#define B_ROWS   4096
#define L_COLS   2048
#define TPB      256
#define NWAVE    (TPB / 32)
#define VPT      (L_COLS / TPB)   // 8 elements per thread
#define INV_TAU  (1.0f / 0.95f)
#define LOG_TAU  (-0.05129329438755058f)   // log(0.95)
#define LOG_CLAMP_F (-100.0f)

// ---- CDNA5 async global->LDS (ASYNCcnt-tracked) ------------------------------
__device__ __forceinline__ void async_b128_to_lds(uint32_t lds_byte_off, uint64_t gaddr) {
    // VDST = VGPR holding LDS byte address, VADDR = 64-bit global address (GV mode)
    asm volatile("global_load_async_to_lds_b128 %0, %1, off"
                 :: "v"(lds_byte_off), "v"(gaddr)
                 : "memory");
}
__device__ __forceinline__ void wait_async_zero() {
    asm volatile("s_wait_asynccnt 0" ::: "memory");
}

__device__ __forceinline__ float bce_term(float p, float y) {
    float lp = fmaxf(logf(p),    LOG_CLAMP_F);
    float lq = fmaxf(log1pf(-p), LOG_CLAMP_F);
    // y*lp + (1-y)*lq
    return fmaf(y, lp - lq, lq);
}

// One workgroup per row: fused cumsum -> F1 -> softmax-weighted sum + 3x BCE partials.
__global__ __launch_bounds__(TPB) void milecut_row_kernel(
    const float* __restrict__ trunc_p,
    const float* __restrict__ v1_p,
    const float* __restrict__ v2_p,
    const float* __restrict__ v3_p,
    const float* __restrict__ labels,
    float* __restrict__ ws)
{
    __shared__ __align__(16) float s_lab[L_COLS];
    __shared__ float s_wtot[NWAVE];
    __shared__ float s_part[NWAVE][5];

    const int row  = blockIdx.x;
    const int tid  = threadIdx.x;
    const int lane = tid & 31;
    const int wid  = tid >> 5;
    const int base = tid * VPT;
    const size_t roff = (size_t)row * (size_t)L_COLS;

    // Async-copy this thread's 8 labels (32 B) into LDS: two B128 async transfers.
    const float* lab_row = labels + roff;
    uint32_t lds0 = (uint32_t)(uintptr_t)(&s_lab[base]);
    async_b128_to_lds(lds0,       (uint64_t)(uintptr_t)(lab_row + base));
    async_b128_to_lds(lds0 + 16u, (uint64_t)(uintptr_t)(lab_row + base + 4));

    // Prefetch the four probability rows while the async engine runs.
    // Locality 3 -> WGP-scope prefetch: pulls lines into all cache levels
    // (SYS-scope would stop at GL2 and not help the consuming WGP).
    __builtin_prefetch(trunc_p + roff + base, 0, 3);
    __builtin_prefetch(v1_p    + roff + base, 0, 3);
    __builtin_prefetch(v2_p    + roff + base, 0, 3);
    __builtin_prefetch(v3_p    + roff + base, 0, 3);

    wait_async_zero();
    __syncthreads();

    // Per-thread serial inclusive prefix over 8 labels (0/1 floats -> exact in fp32).
    float y[VPT], c[VPT];
    float run = 0.0f;
    #pragma unroll
    for (int j = 0; j < VPT; ++j) { y[j] = s_lab[base + j]; run += y[j]; c[j] = run; }
    const float tsum = run;

    // Wave32 inclusive scan of per-thread totals.
    float incl = tsum;
    #pragma unroll
    for (int off = 1; off < 32; off <<= 1) {
        float n = __shfl_up(incl, off, 32);
        if (lane >= off) incl += n;
    }
    if (lane == 31) s_wtot[wid] = incl;
    __syncthreads();

    float woff = 0.0f, rowtot = 0.0f;
    #pragma unroll
    for (int w = 0; w < NWAVE; ++w) {
        float wt = s_wtot[w];
        rowtot += wt;
        if (w < wid) woff += wt;
    }
    const float excl = woff + (incl - tsum);   // exclusive prefix for this thread

    // Vectorized loads of the four probability rows (2x float4 each).
    const float4* tr4 = (const float4*)(trunc_p + roff + base);
    const float4* a4  = (const float4*)(v1_p    + roff + base);
    const float4* b4  = (const float4*)(v2_p    + roff + base);
    const float4* c4  = (const float4*)(v3_p    + roff + base);
    float4 t0 = tr4[0], t1 = tr4[1];
    float4 a0 = a4[0],  a1 = a4[1];
    float4 b0 = b4[0],  b1v = b4[1];
    float4 c0 = c4[0],  c1 = c4[1];
    float tr[VPT] = {t0.x,t0.y,t0.z,t0.w, t1.x,t1.y,t1.z,t1.w};
    float va[VPT] = {a0.x,a0.y,a0.z,a0.w, a1.x,a1.y,a1.z,a1.w};
    float vb[VPT] = {b0.x,b0.y,b0.z,b0.w, b1v.x,b1v.y,b1v.z,b1v.w};
    float vc[VPT] = {c0.x,c0.y,c0.z,c0.w, c1.x,c1.y,c1.z,c1.w};

    float s1 = 0.0f, s2 = 0.0f, bA = 0.0f, bB = 0.0f, bC = 0.0f;
    #pragma unroll
    for (int j = 0; j < VPT; ++j) {
        float cj = excl + c[j];
        float kk = (float)(base + j + 1);
        float r  = (cj > 0.0f) ? (2.0f * cj / (kk + rowtot)) : 0.0f;
        float e  = expf(r * INV_TAU);            // softmax numerator (max-shift cancels in S2/S1)
        s1 += e;
        s2 += e * (logf(tr[j]) - LOG_TAU);       // log(p/tau) = log(p) - log(tau)
        bA += bce_term(va[j], y[j]);
        bB += bce_term(vb[j], y[j]);
        bC += bce_term(vc[j], y[j]);
    }

    // Wave32 butterfly reduction of the 5 partials.
    #pragma unroll
    for (int off = 16; off > 0; off >>= 1) {
        s1 += __shfl_xor(s1, off, 32);
        s2 += __shfl_xor(s2, off, 32);
        bA += __shfl_xor(bA, off, 32);
        bB += __shfl_xor(bB, off, 32);
        bC += __shfl_xor(bC, off, 32);
    }
    if (lane == 0) {
        s_part[wid][0] = s1; s_part[wid][1] = s2;
        s_part[wid][2] = bA; s_part[wid][3] = bB; s_part[wid][4] = bC;
    }
    __syncthreads();
    if (tid == 0) {
        float S1 = 0.0f, S2 = 0.0f, B1 = 0.0f, B2 = 0.0f, B3 = 0.0f;
        #pragma unroll
        for (int w = 0; w < NWAVE; ++w) {
            S1 += s_part[w][0]; S2 += s_part[w][1];
            B1 += s_part[w][2]; B2 += s_part[w][3]; B3 += s_part[w][4];
        }
        ws[0 * B_ROWS + row] = -(S2 / S1);   // row contribution to trunc numerator
        ws[1 * B_ROWS + row] = B1;
        ws[2 * B_ROWS + row] = B2;
        ws[3 * B_ROWS + row] = B3;
    }
}

// Deterministic final reduction of 4 x B_ROWS partials -> scalar loss.
__global__ __launch_bounds__(TPB) void milecut_final_kernel(
    const float* __restrict__ ws, float* __restrict__ out)
{
    __shared__ float red[TPB];
    __shared__ float finals[4];
    const int tid = threadIdx.x;

    for (int a = 0; a < 4; ++a) {
        float local = 0.0f;
        for (int r = tid; r < B_ROWS; r += TPB) local += ws[a * B_ROWS + r];
        red[tid] = local;
        __syncthreads();
        for (int s = TPB / 2; s > 0; s >>= 1) {
            if (tid < s) red[tid] += red[tid + s];
            __syncthreads();
        }
        if (tid == 0) finals[a] = red[0];
        __syncthreads();
    }

    if (tid == 0) {
        float trunc_loss = finals[0] / (float)B_ROWS;                  // -sum(log_out*q)/B
        float denom = (float)B_ROWS * (float)L_COLS * (float)B_ROWS;   // (B*L) * B
        float vsum  = -(finals[1] + finals[2] + finals[3]) / denom;    // v1+v2+v3
        out[0] = 0.5f * trunc_loss + 0.5f * vsum;                      // COEF = 0.5
    }
}

extern "C" void kernel_launch(void* const* d_in, const int* in_sizes, int n_in,
                              void* d_out, int out_size, void* d_ws, size_t ws_size,
                              hipStream_t stream) {
    (void)in_sizes; (void)n_in; (void)out_size; (void)ws_size;
    const float* trunc_p = (const float*)d_in[0];
    const float* v1_p    = (const float*)d_in[1];
    const float* v2_p    = (const float*)d_in[2];
    const float* v3_p    = (const float*)d_in[3];
    const float* labels  = (const float*)d_in[4];
    float* ws  = (float*)d_ws;     // uses 4 * 4096 floats = 64 KB of scratch
    float* out = (float*)d_out;

    milecut_row_kernel<<<B_ROWS, TPB, 0, stream>>>(trunc_p, v1_p, v2_p, v3_p, labels, ws);
    milecut_final_kernel<<<1, TPB, 0, stream>>>(ws, out);
}